// ConstituentAttention_2860448219590
// MI455X (gfx1250) — compile-verified
//
#include <hip/hip_runtime.h>
#include <hip/hip_bf16.h>
#include <math.h>

#define EMB   1024
#define PDIM  64
#define P2    128
#define BB    8
#define SS    2048
#define NBS   (BB * SS)          // 16384 flat (b,s) rows

typedef float v2f __attribute__((ext_vector_type(2)));
typedef float v8f __attribute__((ext_vector_type(8)));

// ---------------------------------------------------------------------------
// K1: qk[r, p] = sum_e context[s, b, e] * W[p, e] + bias[p],  r = b*S + s
// One wave computes one 16(M=rows) x 16(N=p) tile via V_WMMA_F32_16X16X4_F32,
// looping K over E in steps of 4.
// A 16x4 fragment: lanes 0-15 -> {K=0,K=1}, lanes 16-31 -> {K=2,K=3} (b64 load)
// B 4x16 fragment: same split on K, N = lane%16                   (b64 load)
// D: VGPR v -> M = v + (lane<16 ? 0 : 8), N = lane%16
// ---------------------------------------------------------------------------
__global__ void __launch_bounds__(256)
ca_gemm_qk_kernel(const float* __restrict__ ctx,     // (S, B, E)
                  const float* __restrict__ W,       // (2P, E)
                  const float* __restrict__ bias,    // (2P)
                  float* __restrict__ qk) {          // (B*S, 2P)
    const int lane  = threadIdx.x & 31;
    const int wave  = blockIdx.x * (blockDim.x >> 5) + (threadIdx.x >> 5);
    const int tileN = wave & 7;          // 8 tiles over P2=128
    const int tileM = wave >> 3;         // 1024 tiles over NBS=16384
    const int m     = lane & 15;
    const int khalf = (lane >> 4) << 1;  // 0 for lanes 0-15, 2 for lanes 16-31

    const int r = tileM * 16 + m;        // flat row = b*S + s
    const int b = r >> 11;               // / 2048
    const int s = r & (SS - 1);
    const float* __restrict__ arow = ctx + ((size_t)(s * BB + b)) * EMB;
    const int p = tileN * 16 + m;
    const float* __restrict__ brow = W + (size_t)p * EMB;

    v8f c = {};
#pragma unroll 4
    for (int k0 = 0; k0 < EMB; k0 += 4) {
        v2f a  = *(const v2f*)(arow + k0 + khalf);
        v2f bf = *(const v2f*)(brow + k0 + khalf);
        c = __builtin_amdgcn_wmma_f32_16x16x4_f32(
                /*neg_a=*/false, a, /*neg_b=*/false, bf,
                /*c_mod=*/(short)0, c, /*reuse_a=*/false, /*reuse_b=*/false);
    }

    const float bv      = bias[p];
    const int   rowbase = tileM * 16 + ((lane < 16) ? 0 : 8);
    const int   col     = tileN * 16 + m;
#pragma unroll
    for (int v = 0; v < 8; ++v) {
        qk[(size_t)(rowbase + v) * P2 + col] = c[v] + bv;
    }
}

// ---------------------------------------------------------------------------
// K2: per-(b,s): f = dot(q[s], k[s+1])/E, g = dot(q[s], k[s-1])/E,
// 2-way softmax with -inf boundaries -> p0, p1
// ---------------------------------------------------------------------------
__global__ void __launch_bounds__(256)
ca_score_kernel(const float* __restrict__ qk, float* __restrict__ p0,
                float* __restrict__ p1) {
    const int idx = blockIdx.x * blockDim.x + threadIdx.x;  // b*S + s
    if (idx >= NBS) return;
    const int s = idx & (SS - 1);
    const float* __restrict__ q = qk + (size_t)idx * P2;

    float f = 0.f, g = 0.f;
    if (s < SS - 1) {
        const float* __restrict__ kn = qk + (size_t)(idx + 1) * P2 + PDIM;
#pragma unroll 8
        for (int j = 0; j < PDIM; ++j) f += q[j] * kn[j];
        f *= (1.0f / (float)EMB);
    }
    if (s > 0) {
        const float* __restrict__ kp = qk + (size_t)(idx - 1) * P2 + PDIM;
#pragma unroll 8
        for (int j = 0; j < PDIM; ++j) g += q[j] * kp[j];
        g *= (1.0f / (float)EMB);
    }

    float pp0, pp1;
    if (s == 0)            { pp0 = 1.f; pp1 = 0.f; }
    else if (s == SS - 1)  { pp0 = 0.f; pp1 = 1.f; }
    else {
        const float mx  = fmaxf(f, g);
        const float ef  = __expf(f - mx);
        const float eg  = __expf(g - mx);
        const float inv = 1.f / (ef + eg);
        pp0 = ef * inv; pp1 = eg * inv;
    }
    p0[idx] = pp0;
    p1[idx] = pp1;
}

// ---------------------------------------------------------------------------
// K3: flattened roll(-1) of p1, prob0 = sqrt(p0*shift + 1e-6),
// neighbor = prior + (1-prior)*prob0 -> output tail; logp = log(neighbor)
// ---------------------------------------------------------------------------
__global__ void __launch_bounds__(256)
ca_neighbor_kernel(const float* __restrict__ p0, const float* __restrict__ p1,
                   const float* __restrict__ prior,
                   float* __restrict__ out_neighbor,   // d_out + B*S*S
                   float* __restrict__ logp) {
    const int idx = blockIdx.x * blockDim.x + threadIdx.x;
    if (idx >= NBS) return;
    const int nxt   = (idx + 1 == NBS) ? 0 : idx + 1;   // roll over flat (b*s)
    const float shp = p1[nxt];
    const float pr0 = sqrtf(p0[idx] * shp + 1e-6f);
    const float pr  = prior[idx];
    const float na  = pr + (1.f - pr) * pr0;
    out_neighbor[idx] = na;
    logp[idx]         = logf(na);
}

// ---------------------------------------------------------------------------
// K4: per-batch exclusive prefix sum of logp row (2048 elems) -> cs
// Hillis-Steele in LDS, one block (1024 threads) per batch.
// ---------------------------------------------------------------------------
__global__ void __launch_bounds__(1024)
ca_scan_kernel(const float* __restrict__ logp, float* __restrict__ cs) {
    __shared__ float buf[2][SS];
    const int b = blockIdx.x;
    const int t = threadIdx.x;

    for (int i = t; i < SS; i += 1024) buf[0][i] = logp[b * SS + i];
    __syncthreads();

    int src = 0;
    for (int d = 1; d < SS; d <<= 1) {
        const int dst = src ^ 1;
        for (int i = t; i < SS; i += 1024) {
            float v = buf[src][i];
            if (i >= d) v += buf[src][i - d];
            buf[dst][i] = v;
        }
        __syncthreads();
        src = dst;
    }
    // exclusive: cs[0] = 0, cs[i] = inclusive[i-1]  (logp[S-1] never consumed)
    for (int i = t; i < SS; i += 1024)
        cs[b * SS + i] = (i == 0) ? 0.f : buf[src][i - 1];
}

// ---------------------------------------------------------------------------
// K5: constituent_attn[b,i,j] = (i==j) ? 0 : exp(cs[b,max(i,j)] - cs[b,min(i,j)])
// Pure streaming write (134 MB) -> float4 stores, grid (S/1024, S, B).
// ---------------------------------------------------------------------------
__global__ void __launch_bounds__(256)
ca_constituent_kernel(const float* __restrict__ cs, float* __restrict__ out) {
    const int b  = blockIdx.z;
    const int i  = blockIdx.y;
    const int j0 = (blockIdx.x * blockDim.x + threadIdx.x) * 4;

    const float* __restrict__ csb = cs + b * SS;
    const float ci = csb[i];
    const float4 cj = *(const float4*)(csb + j0);

    float4 r;
    {
        const int jj = j0 + 0;
        r.x = (jj == i) ? 0.f : __expf((jj > i) ? (cj.x - ci) : (ci - cj.x));
    }
    {
        const int jj = j0 + 1;
        r.y = (jj == i) ? 0.f : __expf((jj > i) ? (cj.y - ci) : (ci - cj.y));
    }
    {
        const int jj = j0 + 2;
        r.z = (jj == i) ? 0.f : __expf((jj > i) ? (cj.z - ci) : (ci - cj.z));
    }
    {
        const int jj = j0 + 3;
        r.w = (jj == i) ? 0.f : __expf((jj > i) ? (cj.w - ci) : (ci - cj.w));
    }

    float* __restrict__ orow = out + ((size_t)b * SS + i) * SS + j0;
    *(float4*)orow = r;
}

// ---------------------------------------------------------------------------
extern "C" void kernel_launch(void* const* d_in, const int* in_sizes, int n_in,
                              void* d_out, int out_size, void* d_ws, size_t ws_size,
                              hipStream_t stream) {
    const float* ctx   = (const float*)d_in[0];   // (S,B,E)
    const float* prior = (const float*)d_in[1];   // (B,S)
    const float* W     = (const float*)d_in[2];   // (2P,E)
    const float* bias  = (const float*)d_in[3];   // (2P)

    float* out = (float*)d_out;
    float* out_constituent = out;                       // B*S*S
    float* out_neighbor    = out + (size_t)BB * SS * SS;  // B*S

    // workspace layout (floats)
    float* ws   = (float*)d_ws;
    float* qk   = ws;                                   // NBS * 128
    float* p0   = qk + (size_t)NBS * P2;                // NBS
    float* p1   = p0 + NBS;                             // NBS
    float* logp = p1 + NBS;                             // NBS
    float* cs   = logp + NBS;                           // NBS

    // K1: WMMA GEMM. 8192 tiles, 8 waves (256 thr) per block -> 1024 blocks
    ca_gemm_qk_kernel<<<dim3((NBS / 16) * (P2 / 16) / 8), dim3(256), 0, stream>>>(
        ctx, W, bias, qk);

    // K2: neighbor scores + 2-way softmax
    ca_score_kernel<<<dim3(NBS / 256), dim3(256), 0, stream>>>(qk, p0, p1);

    // K3: roll + prob0 + neighbor_attn + log
    ca_neighbor_kernel<<<dim3(NBS / 256), dim3(256), 0, stream>>>(
        p0, p1, prior, out_neighbor, logp);

    // K4: per-batch exclusive scan
    ca_scan_kernel<<<dim3(BB), dim3(1024), 0, stream>>>(logp, cs);

    // K5: streaming write of constituent attention
    ca_constituent_kernel<<<dim3(SS / (256 * 4), SS, BB), dim3(256), 0, stream>>>(
        cs, out_constituent);
}